// RegionCorr_65214783422660
// MI455X (gfx1250) — compile-verified
//
#include <hip/hip_runtime.h>
#include <math.h>

// Problem constants (from reference setup_inputs)
#define NI 128      // n_image
#define NR 36       // n_regions
#define DD 1024     // d
#define NC 640      // n_caption
#define MW 50       // max_w
#define KK 2048     // fused GEMM K = 2*DD
#define EPSV 1e-8f

typedef float v2f __attribute__((ext_vector_type(2)));
typedef float v8f __attribute__((ext_vector_type(8)));

__device__ __forceinline__ float wave_sum(float v) {
#pragma unroll
    for (int off = 16; off > 0; off >>= 1)
        v += __shfl_xor(v, off, 32);
    return v;
}

// Deterministic block-wide sum for 256 threads (8 wave32s).
__device__ __forceinline__ float block_sum(float v, float* sred) {
    v = wave_sum(v);
    const int lane = threadIdx.x & 31;
    const int wid  = threadIdx.x >> 5;
    if (lane == 0) sred[wid] = v;
    __syncthreads();
    float s = 0.f;
#pragma unroll
    for (int i = 0; i < 8; ++i) s += sred[i];
    __syncthreads();
    return s;
}

__device__ __forceinline__ float leaky(float x) { return x > 0.f ? x : 0.1f * x; }

// One block per image. 256 threads, each owns 4 contiguous columns (float4).
// Produces A row: [0:1024) = img_embed, [1024:2048) = img_region_sum.
__global__ void __launch_bounds__(256)
image_prep_kernel(const float* __restrict__ images, float* __restrict__ A) {
    __shared__ float sred[8];
    const int i = blockIdx.x;
    const int t = threadIdx.x;
    const float4* base = (const float4*)(images + (size_t)i * NR * DD);

    float ax = 0.f, ay = 0.f, az = 0.f, aw = 0.f;  // img_region_sum slice
    for (int r = 0; r < NR; ++r) {
        float4 v = base[r * (DD / 4) + t];
        float lx = leaky(v.x), ly = leaky(v.y), lz = leaky(v.z), lw = leaky(v.w);
        float ss = block_sum(lx * lx + ly * ly + lz * lz + lw * lw, sred);
        float rn = 1.0f / (sqrtf(ss) + EPSV);      // l2norm over d per region
        ax += lx * rn; ay += ly * rn; az += lz * rn; aw += lw * rn;
    }
    // img_embed = l2norm(acc / NR):  norm = sqrt(sum(acc^2)) / NR
    float ss = block_sum(ax * ax + ay * ay + az * az + aw * aw, sred);
    const float invR = 1.0f / (float)NR;
    float inv = 1.0f / (sqrtf(ss) * invR + EPSV);
    float sc = invR * inv;

    float* arow = A + (size_t)i * KK;
    float4 e; e.x = ax * sc; e.y = ay * sc; e.z = az * sc; e.w = aw * sc;
    ((float4*)arow)[t] = e;
    float4 s4; s4.x = ax; s4.y = ay; s4.z = az; s4.w = aw;
    ((float4*)(arow + DD))[t] = s4;
}

// One block per caption. Produces B row:
// [0:1024)   = 0.5 * cap_embed
// [1024:2048)= 0.5 * cap_word_sum / (36*len)
__global__ void __launch_bounds__(256)
caption_prep_kernel(const float* __restrict__ captions, const int* __restrict__ cap_lens,
                    float* __restrict__ B) {
    __shared__ float sred[8];
    const int j = blockIdx.x;
    const int t = threadIdx.x;
    const int len = cap_lens[j];                   // uniform across block
    const float4* base = (const float4*)(captions + (size_t)j * MW * DD);

    float rx = 0.f, ry = 0.f, rz = 0.f, rw = 0.f;  // raw masked sum
    float nx = 0.f, ny = 0.f, nz = 0.f, nw = 0.f;  // normalized masked sum
    for (int w = 0; w < len; ++w) {
        float4 v = base[w * (DD / 4) + t];
        rx += v.x; ry += v.y; rz += v.z; rw += v.w;
        float lx = leaky(v.x), ly = leaky(v.y), lz = leaky(v.z), lw = leaky(v.w);
        float ss = block_sum(lx * lx + ly * ly + lz * lz + lw * lw, sred);
        float rn = 1.0f / (sqrtf(ss) + EPSV);
        nx += lx * rn; ny += ly * rn; nz += lz * rn; nw += lw * rn;
    }
    const float lf = (float)len;
    // cap_embed = l2norm(raw/len): norm = sqrt(sum(raw^2)) / len
    float ss = block_sum(rx * rx + ry * ry + rz * rz + rw * rw, sred);
    float inv = 1.0f / (sqrtf(ss) / lf + EPSV);
    float sc = (1.0f / lf) * inv * 0.5f;           // fold final *0.5 here
    float ws = 0.5f / ((float)NR * lf);            // fold 0.5 / (36*len)

    float* brow = B + (size_t)j * KK;
    float4 e; e.x = rx * sc; e.y = ry * sc; e.z = rz * sc; e.w = rw * sc;
    ((float4*)brow)[t] = e;
    float4 s4; s4.x = nx * ws; s4.y = ny * ws; s4.z = nz * ws; s4.w = nw * ws;
    ((float4*)(brow + DD))[t] = s4;
}

// Fused GEMM: out[128,640] = A[128,2048] @ B[640,2048]^T using f32 WMMA 16x16x4.
// Grid: 40 blocks (one per 16-wide N tile) x 8 waves (one per 16-wide M tile).
// A-frag: lane l -> A[M = m0+(l&15)][K = k + 2*(l>>4) + {0,1}]   (float2)
// B-frag: lane l -> B[N = n0+(l&15)][K = k + 2*(l>>4) + {0,1}]   (float2, B row-major per caption)
// C/D: vgpr r, lane l -> out[m0 + r + 8*(l>>4)][n0 + (l&15)]
__global__ void __launch_bounds__(256)
gemm_wmma_kernel(const float* __restrict__ A, const float* __restrict__ B,
                 float* __restrict__ out) {
    const int lane = threadIdx.x & 31;
    const int wave = threadIdx.x >> 5;
    const int m0 = wave * 16;          // 8 M tiles = 128 rows
    const int n0 = blockIdx.x * 16;    // 40 N tiles = 640 cols
    const int half = lane >> 4;        // 0 or 1
    const int idx  = lane & 15;

    const float* pa = A + (size_t)(m0 + idx) * KK + 2 * half;
    const float* pb = B + (size_t)(n0 + idx) * KK + 2 * half;

    v8f acc = {};
#pragma unroll 8
    for (int k = 0; k < KK; k += 4) {
        v2f a = *(const v2f*)(pa + k);
        v2f b = *(const v2f*)(pb + k);
        acc = __builtin_amdgcn_wmma_f32_16x16x4_f32(
            /*neg_a=*/false, a, /*neg_b=*/false, b,
            /*c_mod=*/(short)0, acc, /*reuse_a=*/false, /*reuse_b=*/false);
    }

#pragma unroll
    for (int r = 0; r < 8; ++r) {
        int row = m0 + r + 8 * half;
        out[(size_t)row * NC + n0 + idx] = acc[r];
    }
}

extern "C" void kernel_launch(void* const* d_in, const int* in_sizes, int n_in,
                              void* d_out, int out_size, void* d_ws, size_t ws_size,
                              hipStream_t stream) {
    const float* images   = (const float*)d_in[0];   // 128*36*1024 f32
    const float* captions = (const float*)d_in[1];   // 640*50*1024 f32
    const int*   cap_lens = (const int*)d_in[2];     // 640 i32
    float* out = (float*)d_out;                      // 128*640 f32

    // Workspace layout: A (128 x 2048 f32) then B (640 x 2048 f32)  ~6.3 MB
    float* A = (float*)d_ws;
    float* B = A + (size_t)NI * KK;

    image_prep_kernel<<<dim3(NI), dim3(256), 0, stream>>>(images, A);
    caption_prep_kernel<<<dim3(NC), dim3(256), 0, stream>>>(captions, cap_lens, B);
    gemm_wmma_kernel<<<dim3(NC / 16), dim3(256), 0, stream>>>(A, B, out);
}